// AllAmplitude_42408507081098
// MI455X (gfx1250) — compile-verified
//
#include <hip/hip_runtime.h>
#include <stdint.h>

#define RRES 4
#define NTHREADS 256
#define EPT 2                         // events per thread (independent streams -> VOPD pairing)
#define EVB (NTHREADS * EPT)          // events per block = 512
#define NARR 7

typedef __attribute__((ext_vector_type(4))) unsigned int v4u;
typedef __attribute__((ext_vector_type(8))) int v8i;
typedef __attribute__((ext_vector_type(4))) int v4i;

#if defined(__has_builtin)
#if __has_builtin(__builtin_amdgcn_tensor_load_to_lds) && __has_builtin(__builtin_amdgcn_s_wait_tensorcnt)
#define USE_TDM 1
#endif
#endif
#ifndef USE_TDM
#define USE_TDM 0
#endif

#if USE_TDM
// One TDM DMA: 2D tile (tile_dim0=EVB events, tile_dim1=RRES rows) of a
// row-major (RRES, N) f32 tensor into LDS at byte offset lds_byte.
// D# layout per CDNA5 ISA ch.8; groups 2/3 zero (2-D tensor).
// clang-23 6-arg builtin: (v4u, v8i, v4i, v4i, v8i, i32 cpol).
__device__ __forceinline__ void tdm_load_rows(const float* gsrc, unsigned lds_byte, unsigned n)
{
    unsigned long long ga = (unsigned long long)(uintptr_t)gsrc;
    v4u g0;
    g0.x = 1u;                                            // count=1, user descriptor
    g0.y = lds_byte;                                      // lds_addr (bytes)
    g0.z = (unsigned)ga;                                  // global_addr[31:0]
    g0.w = ((unsigned)(ga >> 32) & 0x01FFFFFFu) | (2u << 30); // global_addr[56:32], type=2
    v8i g1;
    g1[0] = (int)(2u << 16);                              // data_size=2 (4 bytes/elem)
    g1[1] = (int)((n & 0xFFFFu) << 16);                   // tensor_dim0[15:0] (bits 63:48)
    g1[2] = (int)((n >> 16) | ((unsigned)RRES << 16));    // tensor_dim0[31:16] | tensor_dim1[15:0]
    g1[3] = (int)((unsigned)EVB << 16);                   // tensor_dim1[31:16]=0 | tile_dim0=EVB
    g1[4] = RRES;                                         // tile_dim1=4, tile_dim2=0
    g1[5] = (int)n;                                       // tensor_dim0_stride[31:0] = N
    g1[6] = 0;
    g1[7] = 0;
    v4i z4 = {0, 0, 0, 0};
    v8i z8 = {0, 0, 0, 0, 0, 0, 0, 0};
    __builtin_amdgcn_tensor_load_to_lds(g0, g1, z4, z4, z8, 0);
}
#endif

__global__ __launch_bounds__(NTHREADS, 1) void AllAmplitude_42408507081098_kernel(
    const float* __restrict__ alpha1, const float* __restrict__ beta1,
    const float* __restrict__ gamma1, const float* __restrict__ alpha2,
    const float* __restrict__ beta2,  const float* __restrict__ gamma2,
    const float* __restrict__ mInv,
    const float* __restrict__ m0a, const float* __restrict__ g0a,
    const float* __restrict__ coef_r, const float* __restrict__ coef_i,
    float* __restrict__ out, unsigned n)
{
    __shared__ float tile[NARR * RRES * EVB];   // 56 KB of 320 KB/WGP
    const unsigned tid = threadIdx.x;
    const unsigned ev0 = blockIdx.x * EVB;

    const float* bases[NARR] = {alpha1, beta1, gamma1, alpha2, beta2, gamma2, mInv};

#if USE_TDM
    // Wave 0 issues 7 tensor DMAs (per-wave op; EXEC ignored), waits on
    // TENSORcnt, then the workgroup barrier releases the other 7 waves.
    if (tid < 32) {
        const unsigned ldsBase = (unsigned)(uintptr_t)&tile[0];
#pragma unroll
        for (int a = 0; a < NARR; ++a)
            tdm_load_rows(bases[a] + ev0, ldsBase + (unsigned)a * (RRES * EVB * 4u), n);
        __builtin_amdgcn_s_wait_tensorcnt(0);
    }
    __syncthreads();
#else
    {
#pragma unroll
        for (int e = 0; e < EPT; ++e) {
            unsigned idx = ev0 + e * NTHREADS + tid; if (idx >= n) idx = n - 1;
#pragma unroll
            for (int a = 0; a < NARR; ++a)
#pragma unroll
                for (int r = 0; r < RRES; ++r)
                    tile[(a * RRES + r) * EVB + e * NTHREADS + tid] = bases[a][(size_t)r * n + idx];
        }
    }
    __syncthreads();
#endif

    // Hoisted per-resonance uniforms: coef = coef_r e^{i coef_i}, m0^2, m0*g0.
    float creU[RRES], cimU[RRES], m0sq[RRES], mgU[RRES];
#pragma unroll
    for (int r = 0; r < RRES; ++r) {
        const float m0r = m0a[r], g0r = g0a[r];
        float scw, ccw; __sincosf(coef_i[r], &scw, &ccw);
        creU[r] = coef_r[r] * ccw;
        cimU[r] = coef_r[r] * scw;
        m0sq[r] = m0r * m0r;
        mgU[r]  = m0r * g0r;
    }

    float res[EPT];
#pragma unroll
    for (int e = 0; e < EPT; ++e) {
        const unsigned col = EPT * tid + e;   // adjacent lanes -> adjacent b64, bank-conflict free
        float Tre[3][3] = {{0.f}}, Tim[3][3] = {{0.f}};

#pragma unroll
        for (int r = 0; r < RRES; ++r) {
            const float a1 = tile[(0 * RRES + r) * EVB + col];
            const float b1 = tile[(1 * RRES + r) * EVB + col];
            const float c1 = tile[(2 * RRES + r) * EVB + col];
            const float a2 = tile[(3 * RRES + r) * EVB + col];
            const float b2 = tile[(4 * RRES + r) * EVB + col];
            const float c2 = tile[(5 * RRES + r) * EVB + col];
            const float mm = tile[(6 * RRES + r) * EVB + col];

            // Only 5 sincos per (r,n): gamma1 and alpha2 merge into one phase.
            float sa1, ca1; __sincosf(a1, &sa1, &ca1);
            float sb1, cb1; __sincosf(b1, &sb1, &cb1);
            float sth, cth; __sincosf(c1 + a2, &sth, &cth);
            float sb2, cb2; __sincosf(b2, &sb2, &cb2);
            float sg2, cg2; __sincosf(c2, &sg2, &cg2);

            const float IS2 = 0.70710678118654752f;
            float d1[3][3], d2[3][3];   // Wigner d^1, index 0,1,2 <-> m=-1,0,+1
            d1[0][0] = 0.5f * (1.f + cb1); d1[0][1] = -IS2 * sb1; d1[0][2] = 0.5f * (1.f - cb1);
            d1[1][0] =  IS2 * sb1;         d1[1][1] =  cb1;       d1[1][2] = -IS2 * sb1;
            d1[2][0] = 0.5f * (1.f - cb1); d1[2][1] =  IS2 * sb1; d1[2][2] = 0.5f * (1.f + cb1);
            d2[0][0] = 0.5f * (1.f + cb2); d2[0][1] = -IS2 * sb2; d2[0][2] = 0.5f * (1.f - cb2);
            d2[1][0] =  IS2 * sb2;         d2[1][1] =  cb2;       d2[1][2] = -IS2 * sb2;
            d2[2][0] = 0.5f * (1.f - cb2); d2[2][1] =  IS2 * sb2; d2[2][2] = 0.5f * (1.f + cb2);

            // w(r,n) = coef / (m0^2 - m^2 - i m0 g0)
            const float drb = m0sq[r] - mm * mm;
            const float mg  = mgU[r];
            const float inv = __builtin_amdgcn_rcpf(drb * drb + mg * mg);
            const float wre = (creU[r] * drb - cimU[r] * mg) * inv;
            const float wim = (creU[r] * mg + cimU[r] * drb) * inv;

            // P[a] = w * e^{i a alpha1},  Q[d] = e^{i d gamma2}
            float Pre[3], Pim[3];
            Pre[1] = wre;                    Pim[1] = wim;
            Pre[2] = wre * ca1 - wim * sa1;  Pim[2] = wre * sa1 + wim * ca1;
            Pre[0] = wre * ca1 + wim * sa1;  Pim[0] = wim * ca1 - wre * sa1;
            const float Qre[3] = {cg2, 1.f, cg2};
            const float Qim[3] = {-sg2, 0.f, sg2};

#pragma unroll
            for (int a = 0; a < 3; ++a) {
#pragma unroll
                for (int d = 0; d < 3; ++d) {
                    // G[a][d] = sum_x d1[a][x] * e^{i x th} * d2[x][d]
                    const float pm = d1[a][0] * d2[0][d];
                    const float p0 = d1[a][1] * d2[1][d];
                    const float pp = d1[a][2] * d2[2][d];
                    const float gre = p0 + cth * (pp + pm);
                    const float gim = sth * (pp - pm);
                    const float pqre = Pre[a] * Qre[d] - Pim[a] * Qim[d];
                    const float pqim = Pre[a] * Qim[d] + Pim[a] * Qre[d];
                    Tre[a][d] += pqre * gre - pqim * gim;
                    Tim[a][d] += pqre * gim + pqim * gre;
                }
            }
        }

        // Incoherent sum over external helicities: multiplicity 2/3/2 for delta=-1/0/+1.
        float acc = 0.f;
#pragma unroll
        for (int a = 0; a < 3; ++a) {
            acc += 2.f * (Tre[a][0] * Tre[a][0] + Tim[a][0] * Tim[a][0]);
            acc += 3.f * (Tre[a][1] * Tre[a][1] + Tim[a][1] * Tim[a][1]);
            acc += 2.f * (Tre[a][2] * Tre[a][2] + Tim[a][2] * Tim[a][2]);
        }
        res[e] = acc;
    }

    const unsigned idx0 = ev0 + EPT * tid;
    if (idx0 + 1 < n) {
        float2 o; o.x = res[0]; o.y = res[1];
        *(float2*)&out[idx0] = o;               // single b64 store
    } else if (idx0 < n) {
        out[idx0] = res[0];
    }
}

extern "C" void kernel_launch(void* const* d_in, const int* in_sizes, int n_in,
                              void* d_out, int out_size, void* d_ws, size_t ws_size,
                              hipStream_t stream)
{
    const float* alpha1 = (const float*)d_in[0];
    const float* beta1  = (const float*)d_in[1];
    const float* gamma1 = (const float*)d_in[2];
    const float* alpha2 = (const float*)d_in[3];
    const float* beta2  = (const float*)d_in[4];
    const float* gamma2 = (const float*)d_in[5];
    const float* m      = (const float*)d_in[6];
    const float* m0     = (const float*)d_in[7];
    const float* g0     = (const float*)d_in[8];
    const float* coef_r = (const float*)d_in[9];
    const float* coef_i = (const float*)d_in[10];
    float* out = (float*)d_out;

    const unsigned n = (unsigned)(in_sizes[0] / RRES);   // N events
    dim3 block(NTHREADS);
    dim3 grid((n + EVB - 1) / EVB);
    AllAmplitude_42408507081098_kernel<<<grid, block, 0, stream>>>(
        alpha1, beta1, gamma1, alpha2, beta2, gamma2, m,
        m0, g0, coef_r, coef_i, out, n);
}